// WorkflowGraphGenerator_68959994904771
// MI455X (gfx1250) — compile-verified
//
#include <hip/hip_runtime.h>
#include <hip/hip_bf16.h>
#include <math.h>

// Problem constants (from reference): B=32, N=64, IN_DIM=512, HID=256, OUT=128
#define BQ     32
#define NN     64
#define IN_DIM 512
#define HID    256
#define OUTD   128

typedef float v2f __attribute__((ext_vector_type(2)));
typedef float v8f __attribute__((ext_vector_type(8)));

// ---------------------------------------------------------------------------
// Generic (optionally batched) GEMM using native fp32 WMMA:
//   C[b] = act( A[b] (MxK, row-major, lda=K) @ B[b] (KxN, row-major, ldb=N) + bias )
// One wave (32 threads) computes a 16x64 tile: A fragment reused across 4 WMMAs.
// grid = (N/64, M/16, batches). All dims multiples of (16,64,4) in this problem.
// ---------------------------------------------------------------------------
__global__ __launch_bounds__(32)
void gemm_wmma_kernel(const float* __restrict__ A, const float* __restrict__ B,
                      const float* __restrict__ bias, float* __restrict__ C,
                      int N, int K, long long sA, long long sB, long long sC, int act)
{
    const int lane = threadIdx.x & 31;
    const int half = lane >> 4;   // 0: K pair {k,k+1}, 1: {k+2,k+3} (ISA A/B layout)
    const int lid  = lane & 15;
    const int m0 = blockIdx.y * 16;
    const int n0 = blockIdx.x * 64;

    A += (long long)blockIdx.z * sA;
    B += (long long)blockIdx.z * sB;
    C += (long long)blockIdx.z * sC;

    v8f acc0 = {}, acc1 = {}, acc2 = {}, acc3 = {};
    const float* arow = A + (long long)(m0 + lid) * K;

    for (int k = 0; k < K; k += 4) {
        // A 16x4 fragment: lane(lid) holds row m0+lid, cols k+2*half .. +1  (8B aligned)
        v2f a = *(const v2f*)(arow + k + 2 * half);
        const float* bp = B + (long long)(k + 2 * half) * N + n0 + lid;
        v2f b0; b0.x = bp[0];  b0.y = bp[N];
        v2f b1; b1.x = bp[16]; b1.y = bp[N + 16];
        v2f b2; b2.x = bp[32]; b2.y = bp[N + 32];
        v2f b3; b3.x = bp[48]; b3.y = bp[N + 48];
        acc0 = __builtin_amdgcn_wmma_f32_16x16x4_f32(false, a, false, b0, (short)0, acc0, false, false);
        acc1 = __builtin_amdgcn_wmma_f32_16x16x4_f32(false, a, false, b1, (short)0, acc1, false, false);
        acc2 = __builtin_amdgcn_wmma_f32_16x16x4_f32(false, a, false, b2, (short)0, acc2, false, false);
        acc3 = __builtin_amdgcn_wmma_f32_16x16x4_f32(false, a, false, b3, (short)0, acc3, false, false);
    }

    v8f accs[4] = {acc0, acc1, acc2, acc3};
    #pragma unroll
    for (int t = 0; t < 4; ++t) {
        int col = n0 + t * 16 + lid;
        float bv = bias ? bias[col] : 0.0f;
        #pragma unroll
        for (int v = 0; v < 8; ++v) {
            int row = m0 + v + 8 * half;   // C/D layout: VGPR v -> M=v (+8 for hi half)
            float x = accs[t][v] + bv;
            if (act) x = fmaxf(x, 0.0f);
            C[(long long)row * N + col] = x;
        }
    }
}

// ---------------------------------------------------------------------------
// Build symmetric-normalized GCN adjacency per batch.
// M[i][j] = (dep[b,i,j] > 0) && (j < i)   [sigmoid(x)>0.5 <=> x>0]
// fallback chain i-1 -> i if no edges; Ahat = A + I; Anorm = D^-1/2 Ahat D^-1/2
// One 64-thread block per batch; thread i owns row i as a 64-bit mask.
// ---------------------------------------------------------------------------
__global__ __launch_bounds__(64)
void build_anorm_kernel(const float* __restrict__ dep, float* __restrict__ Anorm)
{
    int b = blockIdx.x, i = threadIdx.x;
    const float* drow = dep + ((long long)b * NN + i) * NN;
    unsigned long long row = 0ull;
    for (int j = 0; j < i; ++j)
        if (drow[j] > 0.0f) row |= (1ull << j);

    __shared__ int hasE;
    __shared__ float dinv[NN];
    if (i == 0) hasE = 0;
    __syncthreads();
    if (row != 0ull) atomicOr(&hasE, 1);
    __syncthreads();

    if (!hasE) row = (i > 0) ? (1ull << (i - 1)) : 0ull;  // chain fallback
    row |= (1ull << i);                                   // self loop
    dinv[i] = 1.0f / sqrtf((float)__popcll(row));
    __syncthreads();

    float di = dinv[i];
    float* out = Anorm + ((long long)b * NN + i) * NN;
    for (int j = 0; j < NN; ++j)
        out[j] = ((row >> j) & 1ull) ? di * dinv[j] : 0.0f;
}

// ---------------------------------------------------------------------------
// All-pairs edge MLP: ep[b,i,j] = sigmoid( sum_k relu(hi[b,i,k]+hj[b,j,k]+be1[k]) * we2[k] + be2 )
// One block per (b,i); 64 threads = j. hi-row + be1 and we2 staged in LDS.
// ---------------------------------------------------------------------------
__global__ __launch_bounds__(64)
void edge_mlp_kernel(const float* __restrict__ h_i, const float* __restrict__ h_j,
                     const float* __restrict__ b_e1, const float* __restrict__ W_e2,
                     const float* __restrict__ b_e2, float* __restrict__ ep)
{
    int b = blockIdx.x >> 6;
    int i = blockIdx.x & 63;
    int tid = threadIdx.x;
    __shared__ float s[HID], w2[HID];
    const float* hi = h_i + ((long long)b * NN + i) * HID;
    for (int k = tid; k < HID; k += 64) { s[k] = hi[k] + b_e1[k]; w2[k] = W_e2[k]; }
    __syncthreads();

    int j = tid;
    const float* hj = h_j + ((long long)b * NN + j) * HID;
    float acc = 0.0f;
    for (int k = 0; k < HID; ++k)
        acc += fmaxf(s[k] + hj[k], 0.0f) * w2[k];
    float p = 1.0f / (1.0f + __expf(-(acc + b_e2[0])));
    if (i == j) p = 0.0f;
    ep[((long long)b * NN + i) * NN + j] = p;
}

// ---------------------------------------------------------------------------
// Greedy DAG augmentation (sequential per batch). 64 threads = rows of the
// reachability bitset R (R[a] bit t == "a reaches t"). Bitset Floyd–Warshall
// closure == the reference's repeated-squaring closure for n=64.
// ---------------------------------------------------------------------------
__global__ __launch_bounds__(64)
void dag_kernel(const float* __restrict__ dep, const float* __restrict__ ep,
                float* __restrict__ adj_out)
{
    int b = blockIdx.x, a = threadIdx.x;
    __shared__ unsigned long long R[NN];
    __shared__ unsigned long long ADJ[NN];
    __shared__ unsigned long long sh_Rj;
    __shared__ int sh_cond;

    // ADJ[s] bit t = edge s->t  = M[t][s] = (dep[b,t,s] > 0) && (s < t)
    unsigned long long arow = 0ull;
    for (int t = a + 1; t < NN; ++t)
        if (dep[((long long)b * NN + t) * NN + a] > 0.0f) arow |= (1ull << t);
    ADJ[a] = arow;
    R[a] = arow | (1ull << a);
    __syncthreads();

    // transitive closure
    for (int k = 0; k < NN; ++k) {
        unsigned long long rk = R[k];
        __syncthreads();
        if ((R[a] >> k) & 1ull) R[a] |= rk;
        __syncthreads();
    }

    const float* epb = ep + (long long)b * NN * NN;
    for (int kk = 0; kk < NN * NN; ++kk) {
        int i = kk >> 6, j = kk & 63;
        if (a == 0) {
            bool cond = (i != j) && !((ADJ[i] >> j) & 1ull)
                        && (epb[i * NN + j] > 0.7f) && !((R[j] >> i) & 1ull);
            if (cond) ADJ[i] |= (1ull << j);
            sh_cond = cond ? 1 : 0;
            sh_Rj = R[j];
        }
        __syncthreads();
        if (sh_cond && ((R[a] >> i) & 1ull)) R[a] |= sh_Rj;
        __syncthreads();
    }

    float* o = adj_out + ((long long)b * NN + a) * NN;
    unsigned long long fin = ADJ[a];
    for (int j = 0; j < NN; ++j) o[j] = (float)((fin >> j) & 1ull);
}

// ---------------------------------------------------------------------------
extern "C" void kernel_launch(void* const* d_in, const int* in_sizes, int n_in,
                              void* d_out, int out_size, void* d_ws, size_t ws_size,
                              hipStream_t stream)
{
    const float* SE   = (const float*)d_in[0];   // [32,64,512]
    const float* DEP  = (const float*)d_in[1];   // [32,64,64]
    const float* W_in = (const float*)d_in[2];   const float* b_in = (const float*)d_in[3];
    const float* W_g1 = (const float*)d_in[4];   const float* b_g1 = (const float*)d_in[5];
    const float* W_g2 = (const float*)d_in[6];   const float* b_g2 = (const float*)d_in[7];
    const float* W_g3 = (const float*)d_in[8];   const float* b_g3 = (const float*)d_in[9];
    const float* W_o1 = (const float*)d_in[10];  const float* b_o1 = (const float*)d_in[11];
    const float* W_o2 = (const float*)d_in[12];  const float* b_o2 = (const float*)d_in[13];
    const float* W_e1 = (const float*)d_in[14];  const float* b_e1 = (const float*)d_in[15];
    const float* W_e2 = (const float*)d_in[16];  const float* b_e2 = (const float*)d_in[17];

    const int MB = BQ * NN;  // 2048 rows for the big GEMMs

    // Output layout: node_emb [32,64,128] | edge_probs [32,64,64] | adjacency [32,64,64]
    float* node_emb = (float*)d_out;
    float* ep       = node_emb + (long long)MB * OUTD;
    float* adj      = ep + (long long)BQ * NN * NN;

    // Workspace layout (floats): ~10.5 MB total
    float* ws    = (float*)d_ws;
    float* X     = ws;  ws += (long long)MB * HID;     // 524288
    float* H     = ws;  ws += (long long)MB * HID;
    float* T     = ws;  ws += (long long)MB * HID;
    float* Anorm = ws;  ws += (long long)BQ * NN * NN; // 131072
    float* Hi    = ws;  ws += (long long)MB * HID;
    float* Hj    = ws;  ws += (long long)MB * HID;

    auto gemm = [&](const float* A, const float* B, const float* bias, float* C,
                    int M, int N, int K, int batches,
                    long long sA, long long sB, long long sC, int act) {
        dim3 grid(N / 64, M / 16, batches);
        hipLaunchKernelGGL(gemm_wmma_kernel, grid, dim3(32), 0, stream,
                           A, B, bias, C, N, K, sA, sB, sC, act);
    };

    // 1) input projection: X = SE @ W_in + b_in
    gemm(SE, W_in, b_in, X, MB, HID, IN_DIM, 1, 0, 0, 0, 0);

    // 2) normalized adjacency
    hipLaunchKernelGGL(build_anorm_kernel, dim3(BQ), dim3(64), 0, stream, DEP, Anorm);

    const long long sAn = (long long)NN * NN;   // 4096
    const long long sH  = (long long)NN * HID;  // 16384

    // 3) GCN layer 1: H = relu(Anorm @ (X@W_g1) + b_g1)
    gemm(X, W_g1, nullptr, T, MB, HID, HID, 1, 0, 0, 0, 0);
    gemm(Anorm, T, b_g1, H, NN, HID, NN, BQ, sAn, sH, sH, 1);
    // GCN layer 2: X = relu(Anorm @ (H@W_g2) + b_g2)
    gemm(H, W_g2, nullptr, T, MB, HID, HID, 1, 0, 0, 0, 0);
    gemm(Anorm, T, b_g2, X, NN, HID, NN, BQ, sAn, sH, sH, 1);
    // GCN layer 3: H = Anorm @ (X@W_g3) + b_g3
    gemm(X, W_g3, nullptr, T, MB, HID, HID, 1, 0, 0, 0, 0);
    gemm(Anorm, T, b_g3, H, NN, HID, NN, BQ, sAn, sH, sH, 0);

    // 4) node MLP: node_emb = relu(H@W_o1 + b_o1) @ W_o2 + b_o2
    gemm(H, W_o1, b_o1, X, MB, HID, HID, 1, 0, 0, 0, 1);
    gemm(X, W_o2, b_o2, node_emb, MB, OUTD, HID, 1, 0, 0, 0, 0);

    // 5) edge predictor halves: Hi = node_emb @ W_e1[:128], Hj = node_emb @ W_e1[128:]
    gemm(node_emb, W_e1,                          nullptr, Hi, MB, HID, OUTD, 1, 0, 0, 0, 0);
    gemm(node_emb, W_e1 + (long long)OUTD * HID,  nullptr, Hj, MB, HID, OUTD, 1, 0, 0, 0, 0);

    // 6) all-pairs edge MLP -> edge_probs
    hipLaunchKernelGGL(edge_mlp_kernel, dim3(BQ * NN), dim3(64), 0, stream,
                       Hi, Hj, b_e1, W_e2, b_e2, ep);

    // 7) greedy DAG augmentation -> adjacency
    hipLaunchKernelGGL(dag_kernel, dim3(BQ), dim3(64), 0, stream, DEP, ep, adj);
}